// DeepseekV3Attention_42485816492037
// MI455X (gfx1250) — compile-verified
//
#include <hip/hip_runtime.h>
#include <hip/hip_bf16.h>

typedef __attribute__((ext_vector_type(2))) float v2f;
typedef __attribute__((ext_vector_type(8))) float v8f;

#define HID 7168
#define BATCH 64
#define HEADS 32
#define KLEN 4096
#define Q_LORA 1536
#define KV_LORA 512
#define NOPE 128
#define ROPE 64
#define VDIM 128
#define DTOT 576           // KV_LORA + ROPE
#define HEAD_SIZE 192      // NOPE + ROPE
#define NSPLIT 8
#define KS (KLEN / NSPLIT) // 512 keys per split
#define CH 16              // keys per chunk
#define SKS 580            // padded LDS row stride (floats), 580 % 64 == 4
#define SPS 17             // padded prob-tile stride

__device__ __forceinline__ v8f wmma4(v2f a, v2f b, v8f c) {
  return __builtin_amdgcn_wmma_f32_16x16x4_f32(false, a, false, b, (short)0, c,
                                               false, false);
}

// Async copy 16B global -> LDS (ASYNCcnt-tracked, no VGPR round trip).
// LDS byte offset = low 32 bits of the generic shared-memory address
// (LDS aperture keeps the offset in addr[31:0] per the CDNA5 aperture rules).
__device__ __forceinline__ void async_copy_b128(void* lds_dst,
                                                const void* gsrc) {
  unsigned lds_off = (unsigned)(unsigned long long)lds_dst;
  unsigned long long gaddr = (unsigned long long)gsrc;
  asm volatile("global_load_async_to_lds_b128 %0, %1, off"
               :
               : "v"(lds_off), "v"(gaddr)
               : "memory");
}

__device__ __forceinline__ void async_wait0() {
  asm volatile("s_wait_asynccnt 0" ::: "memory");
}

// ---------------------------------------------------------------------------
// Generic batched GEMM: C[M x N] = A[M x K] * B[K x N], fp32, row-major.
// One wave32 per 16x64 C strip (4 WMMA tiles, A fragment reused 4x).
// Software-pipelined: next K-step fragments are fetched while the current
// 4 WMMAs execute, so s_wait_loadcnt retires under matrix math.
// grid = (N/64, M/16, nbatch); block = 32.
// ---------------------------------------------------------------------------
__global__ __launch_bounds__(32) void wmma_gemm(
    const float* __restrict__ A, long long bsA, int lda,
    const float* __restrict__ B, long long bsB, int ldb,
    float* __restrict__ C, long long bsC, int ldc, int K) {
  const int l = threadIdx.x;
  A += (long long)blockIdx.z * bsA;
  B += (long long)blockIdx.z * bsB;
  C += (long long)blockIdx.z * bsC;
  const int m = blockIdx.y * 16 + (l & 15);
  const int n0 = blockIdx.x * 64 + (l & 15);
  const int koff = (l & 16) ? 2 : 0; // lanes 16-31 carry K+2,K+3 per ISA layout
  const float* arow = A + (size_t)m * lda + koff;
  const float* bp = B + (size_t)koff * ldb + n0;
  v8f c[4];
#pragma unroll
  for (int t = 0; t < 4; ++t)
#pragma unroll
    for (int v = 0; v < 8; ++v) c[t][v] = 0.0f;

  // prologue: fragments for k = 0
  v2f a_cur, b_cur[4];
  a_cur.x = arow[0];
  a_cur.y = arow[1];
#pragma unroll
  for (int t = 0; t < 4; ++t) {
    b_cur[t].x = bp[t * 16];
    b_cur[t].y = bp[(size_t)ldb + t * 16];
  }
#pragma unroll 2
  for (int k = 0; k < K - 4; k += 4) {
    arow += 4;
    bp += (size_t)4 * ldb;
    // fetch next K-step fragments (independent of current WMMAs)
    v2f a_nxt, b_nxt[4];
    a_nxt.x = arow[0];
    a_nxt.y = arow[1];
#pragma unroll
    for (int t = 0; t < 4; ++t) {
      b_nxt[t].x = bp[t * 16];
      b_nxt[t].y = bp[(size_t)ldb + t * 16];
    }
    // compute current K-step
#pragma unroll
    for (int t = 0; t < 4; ++t) c[t] = wmma4(a_cur, b_cur[t], c[t]);
    a_cur = a_nxt;
#pragma unroll
    for (int t = 0; t < 4; ++t) b_cur[t] = b_nxt[t];
  }
  // epilogue: last K-step
#pragma unroll
  for (int t = 0; t < 4; ++t) c[t] = wmma4(a_cur, b_cur[t], c[t]);

  const int rbase = blockIdx.y * 16 + ((l & 16) ? 8 : 0);
#pragma unroll
  for (int t = 0; t < 4; ++t)
#pragma unroll
    for (int v = 0; v < 8; ++v)
      C[(size_t)(rbase + v) * ldc + n0 + t * 16] = c[t][v];
}

// ---------------------------------------------------------------------------
// Row-wise RMSNorm in place: x[row] *= rsqrt(mean(x^2)+eps) * w
// grid = rows, block = 256
// ---------------------------------------------------------------------------
__global__ __launch_bounds__(256) void rmsnorm_rows(float* __restrict__ x,
                                                    const float* __restrict__ w,
                                                    int n) {
  __shared__ float red[256];
  float* p = x + (size_t)blockIdx.x * n;
  float vals[8];
  int cnt = 0;
  float acc = 0.0f;
  for (int i = threadIdx.x; i < n; i += 256) {
    float v = p[i];
    vals[cnt++] = v;
    acc += v * v;
  }
  red[threadIdx.x] = acc;
  __syncthreads();
  for (int s = 128; s > 0; s >>= 1) {
    if (threadIdx.x < s) red[threadIdx.x] += red[threadIdx.x + s];
    __syncthreads();
  }
  float r = rsqrtf(red[0] / (float)n + 1e-6f);
  cnt = 0;
  for (int i = threadIdx.x; i < n; i += 256) p[i] = vals[cnt++] * r * w[i];
}

// ---------------------------------------------------------------------------
// latent_k[b] = concat(rmsnorm(ckv[b,:512], kv_ln_w), rope(ckv[b,512:576]))
// grid = B, block = 256
// ---------------------------------------------------------------------------
__global__ __launch_bounds__(256) void build_latent(
    const float* __restrict__ ckv, const float* __restrict__ kv_ln_w,
    const float* __restrict__ cosb, const float* __restrict__ sinb,
    float* __restrict__ latent) {
  __shared__ float red[256];
  const int b = blockIdx.x;
  const float* src = ckv + (size_t)b * DTOT;
  float* dst = latent + (size_t)b * DTOT;
  float v0 = src[threadIdx.x];
  float v1 = src[threadIdx.x + 256];
  red[threadIdx.x] = v0 * v0 + v1 * v1;
  __syncthreads();
  for (int s = 128; s > 0; s >>= 1) {
    if (threadIdx.x < s) red[threadIdx.x] += red[threadIdx.x + s];
    __syncthreads();
  }
  float r = rsqrtf(red[0] / (float)KV_LORA + 1e-6f);
  dst[threadIdx.x] = v0 * r * kv_ln_w[threadIdx.x];
  dst[threadIdx.x + 256] = v1 * r * kv_ln_w[threadIdx.x + 256];
  if (threadIdx.x < ROPE / 2) {
    int i = threadIdx.x;
    float e = src[KV_LORA + 2 * i];     // x1 = even elements
    float o = src[KV_LORA + 2 * i + 1]; // x2 = odd elements
    float c = cosb[b * (ROPE / 2) + i];
    float s = sinb[b * (ROPE / 2) + i];
    dst[KV_LORA + i] = e * c - o * s;
    dst[KV_LORA + ROPE / 2 + i] = o * c + e * s;
  }
}

// ---------------------------------------------------------------------------
// q_pe rope -> query[b,h,512:576].  grid = B*H blocks, block = 32.
// ---------------------------------------------------------------------------
__global__ __launch_bounds__(32) void qpe_rope(const float* __restrict__ q,
                                               const float* __restrict__ cosb,
                                               const float* __restrict__ sinb,
                                               float* __restrict__ query) {
  const int b = blockIdx.x >> 5;
  const int h = blockIdx.x & 31;
  const int i = threadIdx.x;
  const float* src = q + (size_t)b * (HEADS * HEAD_SIZE) + h * HEAD_SIZE + NOPE;
  float* dst = query + ((size_t)b * HEADS + h) * DTOT;
  float e = src[2 * i];
  float o = src[2 * i + 1];
  float c = cosb[b * (ROPE / 2) + i];
  float s = sinb[b * (ROPE / 2) + i];
  dst[KV_LORA + i] = e * c - o * s;
  dst[KV_LORA + ROPE / 2 + i] = o * c + e * s;
}

// ---------------------------------------------------------------------------
// Flash-decode partial attention: one block per (b, split).
// K/V chunks staged to LDS via global_load_async_to_lds_b128 (ASYNCcnt).
// 8 waves. Waves 0-1: S = Q*K^T (WMMA).  Wave 0: online softmax row update.
// All waves: acc rescale + P*V (WMMA).  Unnormalized partials to workspace.
// ---------------------------------------------------------------------------
__global__ __launch_bounds__(256) void attn_partial(
    const float* __restrict__ query,    // [B][H][576]
    const float* __restrict__ kv_cache, // [B][K][576]
    const float* __restrict__ latent,   // [B][576]  (the updated slot row)
    const int* __restrict__ slots, const int* __restrict__ seq_lens,
    float* __restrict__ part_out,       // [B][NSPLIT][32][512]
    float* __restrict__ part_ml) {      // [B][NSPLIT][32][2]
  __shared__ float sK[CH * SKS];  // key/value chunk, padded rows
  __shared__ float sP[HEADS * SPS];
  __shared__ float sM[HEADS], sL[HEADS], sScale[HEADS];

  const float SM_SCALE = 0.07216878364870322f; // 192^-0.5
  const int b = blockIdx.x, split = blockIdx.y;
  const int tid = threadIdx.x, w = tid >> 5, l = tid & 31;
  const int slot = slots[b];
  const int seqlen = seq_lens[b];
  const float* Q = query + (size_t)b * HEADS * DTOT;
  const int mt = w & 1;           // accumulator m-tile of this wave
  const int ntb = (w >> 1) * 8;   // first of 8 n-tiles of this wave

  if (tid < HEADS) {
    sM[tid] = -__builtin_inff();
    sL[tid] = 0.0f;
  }
  v8f acc[8];
#pragma unroll
  for (int t = 0; t < 8; ++t)
#pragma unroll
    for (int v = 0; v < 8; ++v) acc[t][v] = 0.0f;
  __syncthreads();

  for (int c = 0; c < KS / CH; ++c) {
    const int k0 = split * KS + c * CH;
    // ---- stage 16 x 576 chunk into LDS via async copy (16B granules) ----
    for (int i = tid; i < CH * (DTOT / 4); i += 256) {
      int r = i / (DTOT / 4);
      int d4 = (i % (DTOT / 4)) * 4;
      int kk = k0 + r;
      const float* src = (kk == slot)
                             ? (latent + (size_t)b * DTOT + d4)
                             : (kv_cache + ((size_t)b * KLEN + kk) * DTOT + d4);
      async_copy_b128(&sK[r * SKS + d4], src);
      if (kk + CH < KLEN)  // prefetch next chunk's line (global_prefetch_b8)
        __builtin_prefetch(kv_cache + ((size_t)b * KLEN + kk + CH) * DTOT + d4,
                           0, 0);
    }
    async_wait0();
    __syncthreads();

    // ---- S tiles: waves 0,1 -> 16x16 scores each, K-dim 576 ----
    if (w < 2) {
      v8f s;
#pragma unroll
      for (int v = 0; v < 8; ++v) s[v] = 0.0f;
      const int m = w * 16 + (l & 15);
      const int j = l & 15;
      const int koff = (l & 16) ? 2 : 0;
      const float* qrow = Q + (size_t)m * DTOT;
#pragma unroll 4
      for (int kk = 0; kk < DTOT; kk += 4) {
        v2f a;
        a.x = qrow[kk + koff];
        a.y = qrow[kk + koff + 1];
        v2f bb;                                   // B = K^T: B[d][j] = sK[j][d]
        bb.x = sK[j * SKS + kk + koff];
        bb.y = sK[j * SKS + kk + koff + 1];
        s = wmma4(a, bb, s);
      }
      const int rbase = w * 16 + ((l & 16) ? 8 : 0);
#pragma unroll
      for (int v = 0; v < 8; ++v)
        sP[(rbase + v) * SPS + j] = s[v] * SM_SCALE;
    }
    __syncthreads();

    // ---- online softmax: wave 0, one thread per head-row ----
    if (tid < HEADS) {
      const int r = tid;
      float mOld = sM[r], lOld = sL[r];
      float sc[CH];
      float cmax = -__builtin_inff();
#pragma unroll
      for (int j = 0; j < CH; ++j) {
        bool valid = (k0 + j) < seqlen;
        float x = valid ? sP[r * SPS + j] : -__builtin_inff();
        sc[j] = x;
        cmax = fmaxf(cmax, x);
      }
      float mNew = fmaxf(mOld, cmax);
      float scale = (mOld == -__builtin_inff()) ? 0.0f : __expf(mOld - mNew);
      float lsum = 0.0f;
#pragma unroll
      for (int j = 0; j < CH; ++j) {
        float p = (sc[j] == -__builtin_inff()) ? 0.0f : __expf(sc[j] - mNew);
        sP[r * SPS + j] = p;
        lsum += p;
      }
      sM[r] = mNew;
      sL[r] = lOld * scale + lsum;
      sScale[r] = scale;
    }
    __syncthreads();

    // ---- all waves: rescale accumulators, then P*V (K-dim = 16) ----
    {
      float rs[8];
      const int rbase = mt * 16 + ((l & 16) ? 8 : 0);
#pragma unroll
      for (int v = 0; v < 8; ++v) rs[v] = sScale[rbase + v];
#pragma unroll
      for (int t = 0; t < 8; ++t)
#pragma unroll
        for (int v = 0; v < 8; ++v) acc[t][v] *= rs[v];

      const int m = mt * 16 + (l & 15);
      const int koff = (l & 16) ? 2 : 0;
#pragma unroll
      for (int kk = 0; kk < CH; kk += 4) {
        v2f a;
        a.x = sP[m * SPS + kk + koff];
        a.y = sP[m * SPS + kk + koff + 1];
#pragma unroll
        for (int t = 0; t < 8; ++t) {
          const int n = (ntb + t) * 16 + (l & 15); // V[j][n] = sK[j][n], n<512
          v2f bb;
          bb.x = sK[(kk + koff) * SKS + n];
          bb.y = sK[(kk + koff + 1) * SKS + n];
          acc[t] = wmma4(a, bb, acc[t]);
        }
      }
    }
    __syncthreads();
  }

  // ---- write unnormalized partials + (m, l) ----
  float* po = part_out + (((size_t)b * NSPLIT + split) * HEADS) * KV_LORA;
  const int rbase = mt * 16 + ((l & 16) ? 8 : 0);
#pragma unroll
  for (int t = 0; t < 8; ++t) {
    const int n = (ntb + t) * 16 + (l & 15);
#pragma unroll
    for (int v = 0; v < 8; ++v) po[(size_t)(rbase + v) * KV_LORA + n] = acc[t][v];
  }
  if (tid < HEADS) {
    float* ml = part_ml + (((size_t)b * NSPLIT + split) * HEADS + tid) * 2;
    ml[0] = sM[tid];
    ml[1] = sL[tid];
  }
}

// ---------------------------------------------------------------------------
// Combine split partials -> attn[b,h,0:512].  grid = (B, H), block = 256.
// ---------------------------------------------------------------------------
__global__ __launch_bounds__(256) void attn_reduce(
    const float* __restrict__ part_out, const float* __restrict__ part_ml,
    float* __restrict__ attn) {
  const int b = blockIdx.x, h = blockIdx.y;
  float mv[NSPLIT], lv[NSPLIT], wgt[NSPLIT];
  float M = -__builtin_inff();
#pragma unroll
  for (int s = 0; s < NSPLIT; ++s) {
    const float* ml = part_ml + (((size_t)b * NSPLIT + s) * HEADS + h) * 2;
    mv[s] = ml[0];
    lv[s] = ml[1];
    M = fmaxf(M, mv[s]);
  }
  float L = 0.0f;
#pragma unroll
  for (int s = 0; s < NSPLIT; ++s) {
    wgt[s] = (mv[s] == -__builtin_inff()) ? 0.0f : __expf(mv[s] - M);
    L += wgt[s] * lv[s];
  }
  float invL = (L > 0.0f) ? (1.0f / L) : 0.0f;
  for (int n = threadIdx.x; n < KV_LORA; n += 256) {
    float acc = 0.0f;
#pragma unroll
    for (int s = 0; s < NSPLIT; ++s)
      acc += wgt[s] *
             part_out[((((size_t)b * NSPLIT + s) * HEADS) + h) * KV_LORA + n];
    attn[((size_t)b * HEADS + h) * KV_LORA + n] = acc * invL;
  }
}

// ---------------------------------------------------------------------------
extern "C" void kernel_launch(void* const* d_in, const int* in_sizes, int n_in,
                              void* d_out, int out_size, void* d_ws,
                              size_t ws_size, hipStream_t stream) {
  const float* hidden = (const float*)d_in[0];
  const float* cosb = (const float*)d_in[1];
  const float* sinb = (const float*)d_in[2];
  const float* kv_cache = (const float*)d_in[3];
  const float* w_q_a = (const float*)d_in[4];
  const float* q_a_ln_w = (const float*)d_in[5];
  const float* w_q_b = (const float*)d_in[6];
  const float* w_kv_a = (const float*)d_in[7];
  const float* kv_a_ln_w = (const float*)d_in[8];
  const float* w_uk_t = (const float*)d_in[9];
  const float* w_uv = (const float*)d_in[10];
  const float* w_o = (const float*)d_in[11];
  const int* slots = (const int*)d_in[12];
  const int* seq_lens = (const int*)d_in[13];
  float* out = (float*)d_out;

  float* ws = (float*)d_ws;
  float* q_c = ws;                                     // 64*1536
  float* q = q_c + BATCH * Q_LORA;                     // 64*6144
  float* query = q + BATCH * HEADS * HEAD_SIZE;        // 64*32*576
  float* ckv = query + (size_t)BATCH * HEADS * DTOT;   // 64*576
  float* latent = ckv + BATCH * DTOT;                  // 64*576
  float* attn = latent + BATCH * DTOT;                 // 64*32*512
  float* o_head = attn + (size_t)BATCH * HEADS * KV_LORA; // 64*4096
  float* pout = o_head + BATCH * HEADS * VDIM;         // 64*8*32*512
  float* pml = pout + (size_t)BATCH * NSPLIT * HEADS * KV_LORA; // 64*8*32*2

  // q_c = hidden @ w_q_a ; rmsnorm
  wmma_gemm<<<dim3(Q_LORA / 64, BATCH / 16, 1), 32, 0, stream>>>(
      hidden, 0, HID, w_q_a, 0, Q_LORA, q_c, 0, Q_LORA, HID);
  rmsnorm_rows<<<BATCH, 256, 0, stream>>>(q_c, q_a_ln_w, Q_LORA);
  // q = q_c @ w_q_b
  wmma_gemm<<<dim3(HEADS * HEAD_SIZE / 64, BATCH / 16, 1), 32, 0, stream>>>(
      q_c, 0, Q_LORA, w_q_b, 0, HEADS * HEAD_SIZE, q, 0, HEADS * HEAD_SIZE,
      Q_LORA);
  // ckv = hidden @ w_kv_a ; latent_k = [rmsnorm(kv_c), rope(k_pe)]
  wmma_gemm<<<dim3(DTOT / 64, BATCH / 16, 1), 32, 0, stream>>>(
      hidden, 0, HID, w_kv_a, 0, DTOT, ckv, 0, DTOT, HID);
  build_latent<<<BATCH, 256, 0, stream>>>(ckv, kv_a_ln_w, cosb, sinb, latent);
  // ql_nope: per-head (64 x 128) @ (128 x 512) -> query[:, :, 0:512]
  wmma_gemm<<<dim3(KV_LORA / 64, BATCH / 16, HEADS), 32, 0, stream>>>(
      q, HEAD_SIZE, HEADS * HEAD_SIZE, w_uk_t, (long long)NOPE * KV_LORA,
      KV_LORA, query, DTOT, HEADS * DTOT, NOPE);
  // q_pe rope -> query[:, :, 512:576]
  qpe_rope<<<BATCH * HEADS, 32, 0, stream>>>(q, cosb, sinb, query);
  // fused flash-decode attention (split-K) + combine
  attn_partial<<<dim3(BATCH, NSPLIT), 256, 0, stream>>>(
      query, kv_cache, latent, slots, seq_lens, pout, pml);
  attn_reduce<<<dim3(BATCH, HEADS), 256, 0, stream>>>(pout, pml, attn);
  // per-head (64 x 512) @ (512 x 128) -> o_head[b, h*128+v]
  wmma_gemm<<<dim3(VDIM / 64, BATCH / 16, HEADS), 32, 0, stream>>>(
      attn, KV_LORA, HEADS * KV_LORA, w_uv, (long long)KV_LORA * VDIM, VDIM,
      o_head, VDIM, HEADS * VDIM, KV_LORA);
  // out = o_head @ w_o
  wmma_gemm<<<dim3(HID / 64, BATCH / 16, 1), 32, 0, stream>>>(
      o_head, 0, HEADS * VDIM, w_o, 0, HID, out, 0, HID, HEADS * VDIM);
}